// MapImage2Lidar_609885356380
// MI455X (gfx1250) — compile-verified
//
#include <hip/hip_runtime.h>
#include <cstddef>

// ---------------- problem constants (from reference) ----------------
#define GPTS      20000      // G = 100*100*1 * SAMPLE_NUM(2)
#define NVIEW     12         // B(2) * NV(6)
#define PPATCH    704        // 16*44 patch centers
#define NTILE     44         // PPATCH / 16
#define NCHUNK    5          // g-range split for occupancy
#define CHUNKG    4000       // GPTS / NCHUNK (multiple of 32)
#define EPSF      1e-5f
#define IMG_W_F   704.0f
#define IMG_H_F   256.0f

typedef __attribute__((ext_vector_type(2))) float v2f;
typedef __attribute__((ext_vector_type(8))) float v8f;

// =====================================================================
// Stage 1: project voxel sample points into each view's normalized image
// plane. Per-view layout (4*GPTS floats):
//   XY  = base + 0      : interleaved {x, y} pairs         (WMMA B, K=0,1)
//   SQZ = base + 2*GPTS : interleaved {sq-or-inf, 0} pairs (WMMA B, K=2,3)
// sq = x^2+y^2, or +inf when masked (z<=eps or outside (0,1)^2) so masked
// candidates can never win a strict-< argmin.
// =====================================================================
__global__ void mi2l_project(const float* __restrict__ l2i,
                             const float* __restrict__ aug,
                             float* __restrict__ pts)
{
    int g = blockIdx.x * blockDim.x + threadIdx.x;
    int v = blockIdx.y;                 // uniform per block -> scalar matrix loads
    if (g >= GPTS) return;

    const float* M = l2i + v * 16;
    const float* A = aug + v * 16;

    int ks = g & 1;
    int g0 = g >> 1;
    int ix = g0 / 100;
    int iy = g0 - ix * 100;
    // (c+0.5)/100*(100) - 50  and zline = linspace(-5,3,2)
    float px = (ix + 0.5f) - 50.0f;
    float py = (iy + 0.5f) - 50.0f;
    float pz = ks ? 3.0f : -5.0f;

    float p0 = M[0]  * px + M[1]  * py + M[2]  * pz + M[3];
    float p1 = M[4]  * px + M[5]  * py + M[6]  * pz + M[7];
    float p2 = M[8]  * px + M[9]  * py + M[10] * pz + M[11];
    float p3 = M[12] * px + M[13] * py + M[14] * pz + M[15];

    bool  mask = p2 > EPSF;
    float den  = fmaxf(p2, EPSF);
    float x = p0 / den;
    float y = p1 / den;

    float q0 = A[0] * x + A[1] * y + A[2] + A[3] * p3;
    float q1 = A[4] * x + A[5] * y + A[6] + A[7] * p3;

    float p2x = q0 / IMG_W_F;
    float p2y = q1 / IMG_H_F;

    mask = mask && (p2x > 0.0f) && (p2x < 1.0f) && (p2y > 0.0f) && (p2y < 1.0f);
    float sq = mask ? (p2x * p2x + p2y * p2y) : __builtin_inff();

    float* base = pts + (size_t)v * 4 * GPTS;
    v2f xy;  xy[0] = p2x; xy[1] = p2y;
    v2f sz;  sz[0] = sq;  sz[1] = 0.0f;
    *(v2f*)(base + 2 * (size_t)g)               = xy;   // b64 store
    *(v2f*)(base + 2 * (size_t)GPTS + 2 * (size_t)g) = sz;   // b64 store
}

// =====================================================================
// Stage 2: WMMA-based nearest-point search.
// Score(p, g) = sq_g - 2*pp_p . p2_g   (same argmin as full d2).
// A (16x4, f32): row m = [-2ppx_m, -2ppy_m, 1, 0]
//   lanes 0-15  hold K=0,1  -> {-2ppx, -2ppy}
//   lanes 16-31 hold K=2,3  -> {1, 0}
// B (4x16, f32): col n = [x_g, y_g, sq_g, 0]
//   lanes 0-15  load {x, y}  from XY;  lanes 16-31 load {sq, 0} from SQZ
//   -> a single branchless global_load_b64 per lane per WMMA.
// One wave per (view, 16-patch tile, g-chunk); unrolled 2x (125 iters).
// =====================================================================
__global__ void mi2l_argmin_wmma(const float* __restrict__ pts,
                                 const long long* __restrict__ pc,
                                 const long long* __restrict__ hw,
                                 float* __restrict__ pbest,
                                 int* __restrict__ pidx)
{
    const int lane = threadIdx.x & 31;
    const int wave = blockIdx.x * (blockDim.x >> 5) + (threadIdx.x >> 5);
    const int chunk = wave % NCHUNK;
    const int tv    = wave / NCHUNK;            // [0, NVIEW*NTILE)
    const int v     = tv / NTILE;
    const int tile  = tv - v * NTILE;

    const int  m  = lane & 15;
    const bool hi = lane >= 16;

    const float whx = (float)hw[1];             // 44
    const float why = (float)hw[0];             // 16

    const int  patch = tile * 16 + m;
    const float ppx = ((float)pc[patch * 4 + 2] + 0.5f) / whx;
    const float ppy = ((float)pc[patch * 4 + 3] + 0.5f) / why;

    v2f a;
    a[0] = hi ? 1.0f : (-2.0f * ppx);
    a[1] = hi ? 0.0f : (-2.0f * ppy);

    // per-lane B base: XY pairs for lanes 0-15, SQZ pairs for lanes 16-31,
    // plus this lane's column offset (2*m floats). Loop adds only 2*gb.
    const float* base = pts + (size_t)v * 4 * GPTS;
    const float* bp   = base + (hi ? (size_t)(2 * GPTS) : (size_t)0) + 2 * (size_t)m;

    float best[8];
    int   bidx[8];
#pragma unroll
    for (int r = 0; r < 8; ++r) { best[r] = __builtin_inff(); bidx[r] = 0; }

    v8f cz = {};
    const int gbeg = chunk * CHUNKG;
    const int gend = gbeg + CHUNKG;

    for (int gb = gbeg; gb < gend; gb += 32) {
        // issue both b64 loads before either WMMA so the second load
        // overlaps the first WMMA
        v2f b0 = *(const v2f*)(bp + 2 * (size_t)gb);
        v2f b1 = *(const v2f*)(bp + 2 * (size_t)(gb + 16));
        __builtin_prefetch(bp + 2 * (size_t)(gb + 32), 0, 1);

        v8f d0 = __builtin_amdgcn_wmma_f32_16x16x4_f32(
            false, a, false, b0, (short)0, cz, false, false);
        v8f d1 = __builtin_amdgcn_wmma_f32_16x16x4_f32(
            false, a, false, b1, (short)0, cz, false, false);

        const int gg0 = gb + m;
        const int gg1 = gb + 16 + m;
#pragma unroll
        for (int r = 0; r < 8; ++r) {
            float dv = d0[r];
            bool  lt = dv < best[r];            // strict < -> first-index ties
            best[r] = lt ? dv : best[r];
            bidx[r] = lt ? gg0 : bidx[r];
        }
#pragma unroll
        for (int r = 0; r < 8; ++r) {
            float dv = d1[r];
            bool  lt = dv < best[r];
            best[r] = lt ? dv : best[r];
            bidx[r] = lt ? gg1 : bidx[r];
        }
    }

    // xor-butterfly argmin across the 16 columns of each half-wave
    // (masks < 16 never cross the M=0..7 / M=8..15 halves)
#pragma unroll
    for (int r = 0; r < 8; ++r) {
        float bv = best[r];
        int   bi = bidx[r];
#pragma unroll
        for (int off = 8; off >= 1; off >>= 1) {
            float ov = __shfl_xor(bv, off, 32);
            int   oi = __shfl_xor(bi, off, 32);
            if (ov < bv || (ov == bv && oi < bi)) { bv = ov; bi = oi; }
        }
        best[r] = bv;
        bidx[r] = bi;
    }

    if (m == 0) {                               // lanes 0 and 16 write
#pragma unroll
        for (int r = 0; r < 8; ++r) {
            int pw = tile * 16 + (hi ? 8 : 0) + r;      // C row M
            int o  = (v * PPATCH + pw) * NCHUNK + chunk;
            pbest[o] = best[r];
            pidx[o]  = bidx[r];
        }
    }
}

// =====================================================================
// Stage 3: merge the NCHUNK partials (increasing-g order preserves
// argmin first-index semantics), then emit zyx and nearest_dist.
// =====================================================================
__global__ void mi2l_finalize(const float* __restrict__ pts,
                              const float* __restrict__ pbest,
                              const int* __restrict__ pidx,
                              const long long* __restrict__ pc,
                              const long long* __restrict__ hw,
                              float* __restrict__ out)
{
    int t = blockIdx.x * blockDim.x + threadIdx.x;   // [0, NVIEW*PPATCH)
    if (t >= NVIEW * PPATCH) return;
    int v     = t / PPATCH;
    int patch = t - v * PPATCH;

    float best = __builtin_inff();
    int   bi   = 0;                                  // all-masked -> argmin = 0
#pragma unroll
    for (int c = 0; c < NCHUNK; ++c) {
        float bv = pbest[t * NCHUNK + c];
        int   ii = pidx[t * NCHUNK + c];
        if (bv < best) { best = bv; bi = ii; }
    }

    const float* base = pts + (size_t)v * 4 * GPTS;
    float nbx = base[2 * (size_t)bi];
    float nby = base[2 * (size_t)bi + 1];

    float whx = (float)hw[1];
    float why = (float)hw[0];
    float ppx = ((float)pc[patch * 4 + 2] + 0.5f) / whx;
    float ppy = ((float)pc[patch * 4 + 3] + 0.5f) / why;

    float nd0 = fabsf(ppx - nbx) * whx;
    float nd1 = fabsf(ppy - nby) * why;

    // reconstruct voxel position from flat g index
    int g0 = bi >> 1;
    int ix = g0 / 100;
    int iy = g0 - ix * 100;
    float px = (ix + 0.5f) - 50.0f;
    float py = (iy + 0.5f) - 50.0f;
    // gx = clip((px - x1)/(x2-x1)*sx - 0.5, 0, sx-1), (x2-x1)=sx=100
    float gxv = fminf(fmaxf((px + 50.0f) - 0.5f, 0.0f), 99.0f);
    float gyv = fminf(fmaxf((py + 50.0f) - 0.5f, 0.0f), 99.0f);

    out[t * 3 + 0] = 0.0f;
    out[t * 3 + 1] = gyv;
    out[t * 3 + 2] = gxv;

    float* nd = out + NVIEW * PPATCH * 3;
    nd[t * 2 + 0] = nd0;
    nd[t * 2 + 1] = nd1;
}

// =====================================================================
extern "C" void kernel_launch(void* const* d_in, const int* in_sizes, int n_in,
                              void* d_out, int out_size, void* d_ws, size_t ws_size,
                              hipStream_t stream)
{
    (void)in_sizes; (void)n_in; (void)out_size; (void)ws_size;

    // inputs: [0]=camera_imgs (unused, shapes only), [1]=lidar2image f32,
    //         [2]=img_aug_matrix f32, [3]=patch_coords i64, [4]=hw_shape i64
    const float*     l2i = (const float*)d_in[1];
    const float*     aug = (const float*)d_in[2];
    const long long* pc  = (const long long*)d_in[3];
    const long long* hw  = (const long long*)d_in[4];
    float*           out = (float*)d_out;

    // workspace layout (floats): pts[12*4*20000] | pbest[12*704*5] | pidx[...]
    float* pts   = (float*)d_ws;
    float* pbest = pts + (size_t)NVIEW * 4 * GPTS;
    int*   pidx  = (int*)(pbest + (size_t)NVIEW * PPATCH * NCHUNK);

    mi2l_project<<<dim3((GPTS + 255) / 256, NVIEW), 256, 0, stream>>>(l2i, aug, pts);

    // 12 views * 44 tiles * 5 chunks = 2640 waves; 8 wave32 per 256-thread block
    mi2l_argmin_wmma<<<dim3((NVIEW * NTILE * NCHUNK) / 8), 256, 0, stream>>>(
        pts, pc, hw, pbest, pidx);

    mi2l_finalize<<<dim3((NVIEW * PPATCH) / 256), 256, 0, stream>>>(
        pts, pbest, pidx, pc, hw, out);
}